// RoutingGRUModel_7430293422544
// MI455X (gfx1250) — compile-verified
//
#include <hip/hip_runtime.h>
#include <hip/hip_bf16.h>
#include <math.h>

// Shapes
#define BATCH 256
#define NPTS  512
#define DIM   128
#define BN    (BATCH * NPTS)        // 131072
#define G3    (3 * DIM)             // 384

typedef __attribute__((ext_vector_type(2))) float v2f;
typedef __attribute__((ext_vector_type(8))) float v8f;

__device__ __forceinline__ v8f wmma_f32(v2f a, v2f b, v8f c) {
    // D = A(16x4 f32) x B(4x16 f32) + C(16x16 f32)   -> v_wmma_f32_16x16x4_f32
    return __builtin_amdgcn_wmma_f32_16x16x4_f32(
        /*neg_a=*/false, a, /*neg_b=*/false, b,
        /*c_mod=*/(short)0, c, /*reuse_a=*/false, /*reuse_b=*/false);
}

__device__ __forceinline__ float sigf(float x) {
    return 1.0f / (1.0f + __expf(-x));
}

// ---------------------------------------------------------------------------
// 1) x = coords @ in_proj_w.T + in_proj_b    coords:[BN,2] w:[128,2] -> [BN,128]
// ---------------------------------------------------------------------------
__global__ void k_in_proj(const float* __restrict__ coords,
                          const float* __restrict__ w,
                          const float* __restrict__ b,
                          float* __restrict__ out) {
    int idx = blockIdx.x * blockDim.x + threadIdx.x;   // BN*128 threads
    int d  = idx & (DIM - 1);
    int bn = idx >> 7;
    float c0 = coords[bn * 2 + 0];
    float c1 = coords[bn * 2 + 1];
    out[idx] = fmaf(c0, w[d * 2 + 0], fmaf(c1, w[d * 2 + 1], b[d]));
}

// ---------------------------------------------------------------------------
// 2) WMMA f32 GEMM:  OUT[M, NT*128] = A[M,128] @ W[NT*128,128]^T (+ bias)
//    grid.x = M/64; block = 256 (8 waves). k is the single outer loop with ALL
//    NT*4 accumulators live (NT=3 -> 96 VGPRs); per k: 4 A frags from LDS
//    (each dead after NT uses) + NT B frags from global + NT*4 WMMAs.
//    Bounded register pressure (~120 VGPRs) -> no scratch spills.
// ---------------------------------------------------------------------------
template <int NT, bool HAS_BIAS>
__global__ void k_gemm(const float* __restrict__ A,
                       const float* __restrict__ W,
                       const float* __restrict__ bias,
                       float* __restrict__ out) {
    constexpr int Ncols = NT * 128;                     // 8 waves * 16 cols * NT
    __shared__ float As[64 * 128];                      // 32 KB A tile
    const int row0 = blockIdx.x * 64;

    // cooperative load of the 64x128 A tile (8192 floats = 2048 float4)
    const float4* Ain = (const float4*)(A + (size_t)row0 * DIM);
    float4* As4 = (float4*)As;
    for (int i = threadIdx.x; i < 2048; i += 256) As4[i] = Ain[i];
    __syncthreads();

    const int lane = threadIdx.x & 31;
    const int wave = threadIdx.x >> 5;
    const int half = lane >> 4;           // 0: lanes 0-15, 1: lanes 16-31
    const int ml   = lane & 15;

    v8f acc[NT][4];
    #pragma unroll
    for (int i = 0; i < NT; ++i)
        #pragma unroll
        for (int mt = 0; mt < 4; ++mt) acc[i][mt] = (v8f){};

    // wave handles N-tiles nt = wave + 8*i  ->  W rows n = nt*16 + ml
    const float* wbase = W + (size_t)(wave * 16 + ml) * 128 + 2 * half;
    const float* abase = As + ml * 128 + 2 * half;

    #pragma unroll
    for (int k = 0; k < 32; ++k) {
        v2f a[4];
        #pragma unroll
        for (int mt = 0; mt < 4; ++mt)
            a[mt] = *(const v2f*)(abase + mt * 16 * 128 + 4 * k);
        #pragma unroll
        for (int i = 0; i < NT; ++i) {
            const v2f b = *(const v2f*)(wbase + (size_t)i * 8 * 16 * 128 + 4 * k);
            #pragma unroll
            for (int mt = 0; mt < 4; ++mt)
                acc[i][mt] = wmma_f32(a[mt], b, acc[i][mt]);
        }
    }

    #pragma unroll
    for (int i = 0; i < NT; ++i) {
        const int n = (wave + 8 * i) * 16 + ml;
        const float bv = HAS_BIAS ? bias[n] : 0.0f;
        #pragma unroll
        for (int mt = 0; mt < 4; ++mt) {
            float* orow = out + (size_t)(row0 + mt * 16 + half * 8) * Ncols + n;
            #pragma unroll
            for (int v = 0; v < 8; ++v)
                orow[(size_t)v * Ncols] = acc[i][mt][v] + bv;  // row m = v + 8*half
        }
    }
}

// ---------------------------------------------------------------------------
// 3) Persistent GRU scan for one layer. Recurrence is independent per batch
//    row -> grid.x = 16 workgroups, each owns 16 batch rows, loops t=0..511.
//    Wave w owns hidden columns d = 16w..16w+15. Time-invariant W_hh gate
//    fragments: r/z sets live in registers (128 VGPRs), n-gate block (64 KB)
//    lives in LDS -> ~190 VGPR peak (no spill) and per step only 64
//    ds_load_b64 + 96 back-to-back WMMAs per wave. Next step's gx rows are
//    prefetched (global_prefetch_b8) while the WMMAs run.
// ---------------------------------------------------------------------------
__global__ void k_gru_scan(const float* __restrict__ gx,    // [BN, 384] (+b_ih)
                           const float* __restrict__ w_hh,  // [384,128]
                           const float* __restrict__ b_hh,  // [384]
                           float* __restrict__ Hout) {      // [BN, 128]
    extern __shared__ float smem[];
    float* Wn = smem;                    // [128*128]  n-gate rows 256..383
    float* hb = Wn + 128 * 128;          // [16*128]   h tile
    float* bh = hb + 16 * DIM;           // [384]

    {   // cooperative LDS init
        const float4* src = (const float4*)(w_hh + 256 * 128);
        float4* dst = (float4*)Wn;
        for (int i = threadIdx.x; i < (128 * 128) / 4; i += 256) dst[i] = src[i];
        for (int i = threadIdx.x; i < G3; i += 256) bh[i] = b_hh[i];
        for (int i = threadIdx.x; i < 16 * DIM; i += 256) hb[i] = 0.0f;  // h0=0
    }

    const int lane = threadIdx.x & 31;
    const int wave = threadIdx.x >> 5;
    const int half = lane >> 4;
    const int ml   = lane & 15;
    const int b0   = blockIdx.x * 16;
    const int d    = wave * 16 + ml;     // hidden index this lane produces

    // r/z gate B fragments of W_hh: time-invariant, register-resident
    v2f Br[32], Bz[32];
    {
        const float* wr = w_hh + (size_t)(d)       * 128 + 2 * half;
        const float* wz = w_hh + (size_t)(128 + d) * 128 + 2 * half;
        #pragma unroll
        for (int k = 0; k < 32; ++k) {
            Br[k] = *(const v2f*)(wr + 4 * k);
            Bz[k] = *(const v2f*)(wz + 4 * k);
        }
    }

    const float* ar = hb + ml * 128 + 2 * half;          // A frag base (LDS)
    const float* wn = Wn + (size_t)d * 128 + 2 * half;   // n-gate B frag (LDS)

    for (int t = 0; t < NPTS; ++t) {
        __syncthreads();                 // h tile (and first-iter init) ready

        // prefetch next timestep's gx rows (16 rows x 1536B, 96B stride)
        if (t + 1 < NPTS) {
            const int pr = threadIdx.x >> 4;          // 0..15 (batch row)
            const int pc = (threadIdx.x & 15) * 24;   // 96B stride over 1536B
            __builtin_prefetch(gx + ((size_t)(b0 + pr) * NPTS + (t + 1)) * G3 + pc, 0, 3);
        }

        v8f accr = {}, accz = {}, accn = {};
        #pragma unroll
        for (int k = 0; k < 32; ++k) {
            const v2f a = *(const v2f*)(ar + 4 * k);
            accr = wmma_f32(a, Br[k], accr);
            accz = wmma_f32(a, Bz[k], accz);
            accn = wmma_f32(a, *(const v2f*)(wn + 4 * k), accn);
        }

        float hnew[8];
        const float bhr = bh[d], bhz = bh[128 + d], bhn = bh[256 + d];
        #pragma unroll
        for (int v = 0; v < 8; ++v) {
            const int m = v + 8 * half;
            const size_t gbase = ((size_t)(b0 + m) * NPTS + t) * G3;
            const float gxr = gx[gbase + d];
            const float gxz = gx[gbase + 128 + d];
            const float gxn = gx[gbase + 256 + d];
            const float r  = sigf(gxr + accr[v] + bhr);
            const float z  = sigf(gxz + accz[v] + bhz);
            const float nn = tanhf(gxn + r * (accn[v] + bhn));
            const float hp = hb[m * 128 + d];
            hnew[v] = (1.0f - z) * nn + z * hp;
        }

        __syncthreads();                 // everyone done reading old h
        #pragma unroll
        for (int v = 0; v < 8; ++v) {
            const int m = v + 8 * half;
            hb[m * 128 + d] = hnew[v];
            Hout[((size_t)(b0 + m) * NPTS + t) * DIM + d] = hnew[v];
        }
    }
}

// ---------------------------------------------------------------------------
// 4) graph_emb = mean over t of node_emb          one block per b, 128 threads
// ---------------------------------------------------------------------------
__global__ void k_mean(const float* __restrict__ H, float* __restrict__ gemb) {
    const int b = blockIdx.x, d = threadIdx.x;
    const float* p = H + (size_t)b * NPTS * DIM + d;
    float s = 0.0f;
    for (int t = 0; t < NPTS; ++t) s += p[(size_t)t * DIM];
    gemb[b * DIM + d] = s * (1.0f / (float)NPTS);
}

// ---------------------------------------------------------------------------
// 5) Greedy pointer decode. One workgroup per batch element; that element's
//    keys (512x128 f32 = 256KB) resident in LDS; 512 sequential steps of
//    query GEMV + dot products + argmax / logsumexp LDS reductions.
// ---------------------------------------------------------------------------
__global__ void k_decode(const float* __restrict__ node_emb, // [BN,128]
                         const float* __restrict__ keys,     // [BN,128]
                         const float* __restrict__ gemb,     // [B,128]
                         const float* __restrict__ query_w,  // [128,256]
                         float* __restrict__ act_out,        // [B,N] (as float)
                         float* __restrict__ logp_out,       // [B,N]
                         float* __restrict__ ent_out) {      // [B]
    extern __shared__ float smem[];
    float* K      = smem;                 // 512*128
    float* logits = K + NPTS * DIM;       // 512
    float* q      = logits + NPTS;        // 128
    float* ctx    = q + DIM;              // 256
    float* rv     = ctx + 2 * DIM;        // 256
    int*   ri     = (int*)(rv + 256);     // 256
    int*   vis    = ri + 256;             // 512
    int*   ctrl   = vis + NPTS;           // [0] = prev/action

    const int b = blockIdx.x, tid = threadIdx.x;

    const float4* ks = (const float4*)(keys + (size_t)b * NPTS * DIM);
    float4* kd = (float4*)K;
    for (int i = tid; i < (NPTS * DIM) / 4; i += 256) kd[i] = ks[i];
    for (int i = tid; i < NPTS; i += 256) vis[i] = 0;
    if (tid < DIM) ctx[tid] = gemb[b * DIM + tid];
    if (tid == 0) { ctrl[0] = 0; ent_out[b] = 0.0f; }   // greedy => entropy 0
    __syncthreads();

    const float scale = 0.08838834764831845f;           // 1/sqrt(128)

    for (int t = 0; t < NPTS; ++t) {
        const int prev = ctrl[0];
        if (tid < DIM)
            ctx[DIM + tid] = node_emb[((size_t)b * NPTS + prev) * DIM + tid];
        __syncthreads();

        if (tid < DIM) {                                // query = ctx @ Wq^T
            const float* wr = query_w + (size_t)tid * (2 * DIM);
            float s = 0.0f;
            #pragma unroll 8
            for (int i = 0; i < 2 * DIM; ++i) s = fmaf(wr[i], ctx[i], s);
            q[tid] = s;
        }
        __syncthreads();

        #pragma unroll
        for (int rep = 0; rep < 2; ++rep) {             // logits (2 nodes/thread)
            const int n = tid + rep * 256;
            const float* kr = K + n * DIM;
            float s = 0.0f;
            #pragma unroll 8
            for (int i = 0; i < DIM; ++i) s = fmaf(q[i], kr[i], s);
            logits[n] = vis[n] ? -INFINITY : s * scale;
        }
        __syncthreads();

        {   // masked argmax, first-occurrence tie-break (matches jnp.argmax)
            float v0 = logits[tid], v1 = logits[tid + 256];
            float bv; int bi;
            if (v1 > v0) { bv = v1; bi = tid + 256; } else { bv = v0; bi = tid; }
            rv[tid] = bv; ri[tid] = bi;
        }
        __syncthreads();
        for (int off = 128; off > 0; off >>= 1) {
            if (tid < off) {
                const float v2 = rv[tid + off]; const int i2 = ri[tid + off];
                if (v2 > rv[tid] || (v2 == rv[tid] && i2 < ri[tid])) {
                    rv[tid] = v2; ri[tid] = i2;
                }
            }
            __syncthreads();
        }
        const float maxv = rv[0];
        const int action = ri[0];
        __syncthreads();                                 // before rv reuse

        // logsumexp (masked entries: exp(-inf) = 0); logp[action] = -log(sum)
        rv[tid] = __expf(logits[tid] - maxv) + __expf(logits[tid + 256] - maxv);
        __syncthreads();
        for (int off = 128; off > 0; off >>= 1) {
            if (tid < off) rv[tid] += rv[tid + off];
            __syncthreads();
        }
        if (tid == 0) {
            act_out[(size_t)b * NPTS + t]  = (float)action;
            logp_out[(size_t)b * NPTS + t] = -__logf(rv[0]);
            vis[action] = 1;
            ctrl[0] = action;
        }
        __syncthreads();
    }
}

// ---------------------------------------------------------------------------
extern "C" void kernel_launch(void* const* d_in, const int* in_sizes, int n_in,
                              void* d_out, int out_size, void* d_ws, size_t ws_size,
                              hipStream_t stream) {
    const float* coords    = (const float*)d_in[0];   // [B,N,2]
    const float* in_proj_w = (const float*)d_in[1];   // [128,2]
    const float* in_proj_b = (const float*)d_in[2];   // [128]
    const float* gru_w_ih  = (const float*)d_in[3];   // [3,384,128]
    const float* gru_w_hh  = (const float*)d_in[4];   // [3,384,128]
    const float* gru_b_ih  = (const float*)d_in[5];   // [3,384]
    const float* gru_b_hh  = (const float*)d_in[6];   // [3,384]
    const float* query_w   = (const float*)d_in[7];   // [128,256]
    const float* key_w     = (const float*)d_in[8];   // [128,128]

    float* out     = (float*)d_out;
    float* actions = out;                  // [B*N]
    float* logps   = out + BN;             // [B*N]
    float* ent     = out + 2 * BN;         // [B]

    // workspace layout (floats): bufA | bufB | gx | keys | gemb  (~403 MB)
    float* ws      = (float*)d_ws;
    float* bufA    = ws;
    float* bufB    = bufA + (size_t)BN * DIM;
    float* gxbuf   = bufB + (size_t)BN * DIM;
    float* keysbuf = gxbuf + (size_t)BN * G3;
    float* gembbuf = keysbuf + (size_t)BN * DIM;

    // 1) input projection
    k_in_proj<<<(BN * DIM) / 256, 256, 0, stream>>>(coords, in_proj_w, in_proj_b, bufA);

    // 2) 3 GRU layers: WMMA gate precompute + persistent WMMA scan
    const size_t scan_lds = (size_t)(128 * 128 + 16 * DIM + G3) * sizeof(float); // ~74 KB
    float* hin = bufA;
    float* hout = bufB;
    for (int l = 0; l < 3; ++l) {
        k_gemm<3, true><<<BN / 64, 256, 0, stream>>>(
            hin, gru_w_ih + (size_t)l * G3 * DIM, gru_b_ih + l * G3, gxbuf);
        k_gru_scan<<<BATCH / 16, 256, scan_lds, stream>>>(
            gxbuf, gru_w_hh + (size_t)l * G3 * DIM, gru_b_hh + l * G3, hout);
        float* tmp = hin; hin = hout; hout = tmp;
    }
    float* node_emb = hin;

    // 3) keys GEMM + graph embedding
    k_gemm<1, false><<<BN / 64, 256, 0, stream>>>(node_emb, key_w, nullptr, keysbuf);
    k_mean<<<BATCH, DIM, 0, stream>>>(node_emb, gembbuf);

    // 4) greedy decode (keys in LDS, ~270 KB)
    const size_t dec_lds = (size_t)(NPTS * DIM + NPTS + DIM + 2 * DIM + 256) * sizeof(float)
                         + (size_t)(256 + NPTS + 4) * sizeof(int);
    k_decode<<<BATCH, 256, dec_lds, stream>>>(
        node_emb, keysbuf, gembbuf, query_w, actions, logps, ent);
}